// CNN_BiGRU_5823975654003
// MI455X (gfx1250) — compile-verified
//
#include <hip/hip_runtime.h>
#include <math.h>
#include <stdint.h>

// Problem constants (match reference)
#define B_  128
#define S_  256
#define D_  768
#define H_  384
#define CK_ 256
#define M_  (B_*S_)      // 32768 rows of (b,s)
#define EPS_ 1e-5f

typedef __attribute__((ext_vector_type(2))) float v2f;
typedef __attribute__((ext_vector_type(8))) float v8f;

// D(16x16,f32) = A(16x4,f32) x B(4x16,f32) + C  — full-precision f32 WMMA.
// Per-lane fragment layout (wave32):
//   A: lanes 0-15 row M=ln hold K={k0,k0+1}; lanes 16-31 hold K={k0+2,k0+3}
//   B: lanes 0-15 col N=ln hold K rows {k0,k0+1}; lanes 16-31 {k0+2,k0+3}
//   => for Y = X @ W^T with W stored (N,K) row-major, both fragments load
//      [row][k0 + 2*half] , [row][k0 + 2*half + 1]  (row = m for A, n for B)
//   C/D: VGPR r -> (M = r + 8*half, N = ln)
__device__ __forceinline__ v8f wmma4(v2f a, v2f b, v8f c) {
  return __builtin_amdgcn_wmma_f32_16x16x4_f32(false, a, false, b, (short)0, c,
                                               false, false);
}

__device__ __forceinline__ v8f zero8() {
  v8f z;
#pragma unroll
  for (int i = 0; i < 8; ++i) z[i] = 0.f;
  return z;
}

// Async global->LDS copy of 16 bytes (CDNA5: tracked by ASYNCcnt, no VGPR
// staging).  lds byte-offset = low 32 bits of the generic pointer (flat->LDS
// mapping truncates to addr[31:0]).
__device__ __forceinline__ void async_ld_b128(void* lds_dst, const void* gsrc) {
  unsigned lds = (unsigned)(uintptr_t)lds_dst;
  asm volatile("global_load_async_to_lds_b128 %0, %1, off"
               :: "v"(lds), "v"(gsrc) : "memory");
}
__device__ __forceinline__ void wait_async0() {
  asm volatile("s_wait_asynccnt 0x0" ::: "memory");
}

// ---------------------------------------------------------------------------
// Weight repack for the conv branches: w[oc][ic][t] -> wp[t][oc][ic]
// so B-fragments become contiguous 8-byte loads.
// ---------------------------------------------------------------------------
__global__ __launch_bounds__(256)
void repack_kernel(const float* __restrict__ w, float* __restrict__ wp, int KS)
{
  int idx = blockIdx.x * 256 + threadIdx.x;
  int total = CK_ * D_ * KS;
  if (idx >= total) return;
  int t  = idx % KS;
  int r  = idx / KS;
  int ic = r % D_;
  int n  = r / D_;
  wp[((long)t * CK_ + n) * D_ + ic] = w[idx];
}

// ---------------------------------------------------------------------------
// Conv1d branch as tap-wise GEMM with async-LDS staged activations.
// Block = 128 threads (4 waves) handles one 16-row M-tile x 256 oc.
// LDS holds 22 rows (3-row halo each side, zero-filled at sequence edges;
// a 16-row tile never spans batch elements since 16 | S).
// All taps read A-fragments from LDS; B streams from repacked weights in L2.
// ---------------------------------------------------------------------------
#define CXS (D_ + 4)   // padded LDS row stride (dwords)

__global__ __launch_bounds__(128)
void conv_kernel(const float* __restrict__ x, const float* __restrict__ wp,
                 const float* __restrict__ bias, float* __restrict__ y,
                 int KS, int oc_off)
{
  __shared__ float xs[22 * CXS];           // ~67.9 KB

  const int lane = threadIdx.x & 31;
  const int wv   = threadIdx.x >> 5;       // 0..3 -> oc group
  const int half = lane >> 4;
  const int ln   = lane & 15;

  const int m0 = blockIdx.x * 16;
  const int bb = m0 / S_;
  const int s0 = m0 % S_;
  const int P  = KS >> 1;

  // stage rows s0-3 .. s0+18 (16B packets), zero-fill outside [0,S)
  for (int q = threadIdx.x; q < 22 * (D_ / 4); q += blockDim.x) {
    int row = q / (D_ / 4);
    int col = (q % (D_ / 4)) * 4;
    int srel = s0 + row - 3;
    float* dst = &xs[row * CXS + col];
    if (srel >= 0 && srel < S_) {
      async_ld_b128(dst, x + ((long)bb * S_ + srel) * D_ + col);
    } else {
      dst[0] = 0.f; dst[1] = 0.f; dst[2] = 0.f; dst[3] = 0.f;
    }
  }
  wait_async0();
  __syncthreads();

  const int n0 = wv * 64;
  v8f acc[4];
#pragma unroll
  for (int j = 0; j < 4; ++j) acc[j] = zero8();

  for (int t = 0; t < KS; ++t) {
    const float* ar = &xs[(ln + t - P + 3) * CXS];       // staged A row
    const float* wt = wp + (long)t * CK_ * D_;
    for (int k0 = 0; k0 < D_; k0 += 4) {
      int kk = k0 + 2 * half;
      v2f a = *(const v2f*)(ar + kk);
#pragma unroll
      for (int j = 0; j < 4; ++j) {
        int n = n0 + j * 16 + ln;
        v2f bf = *(const v2f*)(wt + (long)n * D_ + kk);
        acc[j] = wmma4(a, bf, acc[j]);
      }
    }
  }

#pragma unroll
  for (int j = 0; j < 4; ++j) {
    int n = n0 + j * 16 + ln;
    float bv = bias[n];
#pragma unroll
    for (int r = 0; r < 8; ++r) {
      int row = m0 + r + 8 * half;
      y[(long)row * D_ + oc_off + n] = acc[j][r] + bv;
    }
  }
}

// ---------------------------------------------------------------------------
// Projection GEMM with async-LDS staged activations:
//   Y[m,n] = sum_k X[m,k]*W[n,k] + bias[n],  K = 768.
// Block = (N/64) waves, one 16-row M-tile; X tile staged once via async b128.
// grid = M_/16 blocks.
// ---------------------------------------------------------------------------
#define PXS (D_ + 4)

__global__ __launch_bounds__(384)
void proj_kernel(const float* __restrict__ X, const float* __restrict__ W,
                 const float* __restrict__ bias, float* __restrict__ Y, int N)
{
  __shared__ float xs[16 * PXS];           // ~49.4 KB

  const int lane = threadIdx.x & 31;
  const int wv   = threadIdx.x >> 5;       // 0 .. N/64-1
  const int half = lane >> 4;
  const int ln   = lane & 15;
  const int m0   = blockIdx.x * 16;

  for (int q = threadIdx.x; q < 16 * (D_ / 4); q += blockDim.x) {
    int row = q / (D_ / 4);
    int col = (q % (D_ / 4)) * 4;
    async_ld_b128(&xs[row * PXS + col], X + (long)(m0 + row) * D_ + col);
  }
  wait_async0();
  __syncthreads();

  const int n0 = wv * 64;
  v8f acc[4];
#pragma unroll
  for (int j = 0; j < 4; ++j) acc[j] = zero8();

  for (int k0 = 0; k0 < D_; k0 += 4) {
    int kk = k0 + 2 * half;
    v2f a = *(const v2f*)(&xs[ln * PXS + kk]);
#pragma unroll
    for (int j = 0; j < 4; ++j) {
      int n = n0 + j * 16 + ln;
      v2f bf = *(const v2f*)(W + (long)n * D_ + kk);
      acc[j] = wmma4(a, bf, acc[j]);
    }
  }

#pragma unroll
  for (int j = 0; j < 4; ++j) {
    int n = n0 + j * 16 + ln;
    float bv = bias[n];
#pragma unroll
    for (int r = 0; r < 8; ++r) {
      int row = m0 + r + 8 * half;
      Y[(long)row * N + n] = acc[j][r] + bv;
    }
  }
}

// ---------------------------------------------------------------------------
// Row LayerNorm: out[row,:] = (in[row,:] - mu)*rsqrt(var+eps)*g + b
// One block (256 threads) per row; in-place safe.
// ---------------------------------------------------------------------------
__global__ __launch_bounds__(256)
void ln_rows(const float* __restrict__ in, float* __restrict__ out,
             const float* __restrict__ g, const float* __restrict__ bta,
             int width)
{
  int row = blockIdx.x;
  const float* p = in + (long)row * width;
  float s = 0.f, ss = 0.f;
  for (int i = threadIdx.x; i < width; i += blockDim.x) {
    float v = p[i]; s += v; ss += v * v;
  }
#pragma unroll
  for (int off = 16; off; off >>= 1) {
    s  += __shfl_xor(s,  off, 32);
    ss += __shfl_xor(ss, off, 32);
  }
  __shared__ float rs[2];
  if (threadIdx.x == 0) { rs[0] = 0.f; rs[1] = 0.f; }
  __syncthreads();
  if ((threadIdx.x & 31) == 0) { atomicAdd(&rs[0], s); atomicAdd(&rs[1], ss); }
  __syncthreads();
  float mu  = rs[0] / (float)width;
  float var = rs[1] / (float)width - mu * mu;
  float inv = rsqrtf(var + EPS_);
  float* q = out + (long)row * width;
  for (int i = threadIdx.x; i < width; i += blockDim.x)
    q[i] = (p[i] - mu) * inv * g[i] + bta[i];
}

// ---------------------------------------------------------------------------
// Bidirectional layer-normalized GRU scan.
// grid (8,2): blockIdx.x = batch group (16 rows), blockIdx.y = direction.
// Block = 384 threads = 12 waves. h (16x384) lives in LDS.
// ---------------------------------------------------------------------------
#define HSS (H_ + 4)   // padded LDS stride for h (avoids 16-way bank conflict)

__global__ __launch_bounds__(384)
void scan_kernel(const float* __restrict__ arz, const float* __restrict__ an,
                 const float* __restrict__ Whrz, const float* __restrict__ Whrz_b,
                 const float* __restrict__ Whn,  const float* __restrict__ Whn_b,
                 const float* __restrict__ lnh1_g, const float* __restrict__ lnh1_b,
                 const float* __restrict__ lnh2_g, const float* __restrict__ lnh2_b,
                 float* __restrict__ outbuf)
{
  __shared__ float h_s[16 * HSS];       // ~24.3 KB
  __shared__ float gates[16 * 768];     // 48 KB  (r | z)
  __shared__ float srz_sum[16], srz_ssq[16], sn_sum[16], sn_ssq[16];

  const int lane = threadIdx.x & 31;
  const int wv   = threadIdx.x >> 5;    // 0..11
  const int half = lane >> 4;
  const int ln   = lane & 15;
  const int dir  = blockIdx.y;
  const int b0   = blockIdx.x * 16;

  for (int i = threadIdx.x; i < 16 * HSS; i += blockDim.x) h_s[i] = 0.f;
  __syncthreads();

  for (int step = 0; step < S_; ++step) {
    int s = dir ? (S_ - 1 - step) : step;

    if (threadIdx.x < 16) {
      srz_sum[threadIdx.x] = 0.f; srz_ssq[threadIdx.x] = 0.f;
      sn_sum[threadIdx.x]  = 0.f; sn_ssq[threadIdx.x]  = 0.f;
    }
    __syncthreads();

    // ---- recurrent GEMMs (WMMA f32): 4 rz tiles + 2 n tiles per wave ----
    v8f accz[4], accn[2];
#pragma unroll
    for (int j = 0; j < 4; ++j) accz[j] = zero8();
#pragma unroll
    for (int j = 0; j < 2; ++j) accn[j] = zero8();

    for (int k0 = 0; k0 < H_; k0 += 4) {
      int kk = k0 + 2 * half;
      v2f a = *(const v2f*)(h_s + ln * HSS + kk);
#pragma unroll
      for (int j = 0; j < 4; ++j) {
        int n = wv * 64 + j * 16 + ln;
        v2f bf = *(const v2f*)(Whrz + (long)n * H_ + kk);
        accz[j] = wmma4(a, bf, accz[j]);
      }
#pragma unroll
      for (int j = 0; j < 2; ++j) {
        int n = wv * 32 + j * 16 + ln;
        v2f bf = *(const v2f*)(Whn + (long)n * H_ + kk);
        accn[j] = wmma4(a, bf, accn[j]);
      }
    }

    // ---- add biases, accumulate per-row LN stats (shfl + LDS atomics) ----
#pragma unroll
    for (int j = 0; j < 4; ++j) {
      int n = wv * 64 + j * 16 + ln;
      float bv = Whrz_b[n];
#pragma unroll
      for (int r = 0; r < 8; ++r) {
        float v = accz[j][r] + bv; accz[j][r] = v;
        float su = v, sq = v * v;
#pragma unroll
        for (int off = 8; off; off >>= 1) {
          su += __shfl_xor(su, off, 32); sq += __shfl_xor(sq, off, 32);
        }
        if (ln == 0) {
          int row = r + 8 * half;
          atomicAdd(&srz_sum[row], su); atomicAdd(&srz_ssq[row], sq);
        }
      }
    }
#pragma unroll
    for (int j = 0; j < 2; ++j) {
      int n = wv * 32 + j * 16 + ln;
      float bv = Whn_b[n];
#pragma unroll
      for (int r = 0; r < 8; ++r) {
        float v = accn[j][r] + bv; accn[j][r] = v;
        float su = v, sq = v * v;
#pragma unroll
        for (int off = 8; off; off >>= 1) {
          su += __shfl_xor(su, off, 32); sq += __shfl_xor(sq, off, 32);
        }
        if (ln == 0) {
          int row = r + 8 * half;
          atomicAdd(&sn_sum[row], su); atomicAdd(&sn_ssq[row], sq);
        }
      }
    }
    __syncthreads();

    // ---- rz: LayerNorm(lnh1) + a_rz + sigmoid -> gates in LDS ----
#pragma unroll
    for (int j = 0; j < 4; ++j) {
      int col = wv * 64 + j * 16 + ln;
      float gg = lnh1_g[col], gb = lnh1_b[col];
#pragma unroll
      for (int r = 0; r < 8; ++r) {
        int row = r + 8 * half;
        float mu  = srz_sum[row] * (1.f / 768.f);
        float var = srz_ssq[row] * (1.f / 768.f) - mu * mu;
        float nv  = (accz[j][r] - mu) * rsqrtf(var + EPS_) * gg + gb;
        long  m   = (long)(b0 + row) * S_ + s;
        float pre = arz[m * 768 + col] + nv;
        gates[row * 768 + col] = 1.f / (1.f + expf(-pre));
      }
    }
    __syncthreads();

    // ---- n: LayerNorm(lnh2), n = tanh(a_n + r*ln), h = (1-z)n + z h ----
#pragma unroll
    for (int j = 0; j < 2; ++j) {
      int col = wv * 32 + j * 16 + ln;
      float gg = lnh2_g[col], gb = lnh2_b[col];
#pragma unroll
      for (int r = 0; r < 8; ++r) {
        int row = r + 8 * half;
        float mu  = sn_sum[row] * (1.f / 384.f);
        float var = sn_ssq[row] * (1.f / 384.f) - mu * mu;
        float nv  = (accn[j][r] - mu) * rsqrtf(var + EPS_) * gg + gb;
        float rg  = gates[row * 768 + col];
        float zg  = gates[row * 768 + 384 + col];
        long  m   = (long)(b0 + row) * S_ + s;
        float nval = tanhf(an[m * H_ + col] + rg * nv);
        float hold = h_s[row * HSS + col];
        float hnew = (1.f - zg) * nval + zg * hold;
        h_s[row * HSS + col] = hnew;
        outbuf[m * 768 + dir * H_ + col] = hnew;   // concat(fwd, bwd)
      }
    }
    __syncthreads();
  }
}

// ---------------------------------------------------------------------------
extern "C" void kernel_launch(void* const* d_in, const int* in_sizes, int n_in,
                              void* d_out, int out_size, void* d_ws, size_t ws_size,
                              hipStream_t stream) {
  const float* x       = (const float*)d_in[0];
  const float* conv_w3 = (const float*)d_in[1];
  const float* conv_b3 = (const float*)d_in[2];
  const float* conv_w5 = (const float*)d_in[3];
  const float* conv_b5 = (const float*)d_in[4];
  const float* conv_w7 = (const float*)d_in[5];
  const float* conv_b7 = (const float*)d_in[6];
  const float* Wxrz_w  = (const float*)d_in[7];
  const float* Wxrz_b  = (const float*)d_in[8];
  const float* Whrz_w  = (const float*)d_in[9];
  const float* Whrz_b  = (const float*)d_in[10];
  const float* Wxn_w   = (const float*)d_in[11];
  const float* Wxn_b   = (const float*)d_in[12];
  const float* Whn_w   = (const float*)d_in[13];
  const float* Whn_b   = (const float*)d_in[14];
  const float* lnx1_g  = (const float*)d_in[15];
  const float* lnx1_b  = (const float*)d_in[16];
  const float* lnh1_g  = (const float*)d_in[17];
  const float* lnh1_b  = (const float*)d_in[18];
  const float* lnx2_g  = (const float*)d_in[19];
  const float* lnx2_b  = (const float*)d_in[20];
  const float* lnh2_g  = (const float*)d_in[21];
  const float* lnh2_b  = (const float*)d_in[22];
  const float* out_ln_g = (const float*)d_in[23];
  const float* out_ln_b = (const float*)d_in[24];
  float* out = (float*)d_out;

  // workspace (floats): cnn(M*768) | arz(M*768) | an(M*384) | wp3 | wp5 | wp7
  float* ws  = (float*)d_ws;
  float* cnn = ws;
  float* arz = cnn + (size_t)M_ * 768;
  float* an  = arz + (size_t)M_ * 768;
  float* wp3 = an  + (size_t)M_ * 384;
  float* wp5 = wp3 + (size_t)CK_ * D_ * 3;
  float* wp7 = wp5 + (size_t)CK_ * D_ * 5;
  float* outc = cnn;   // reuse: cnn_out dead after projections

  // 0) repack conv weights: w[oc][ic][t] -> wp[t][oc][ic]
  repack_kernel<<<(CK_*D_*3 + 255)/256, 256, 0, stream>>>(conv_w3, wp3, 3);
  repack_kernel<<<(CK_*D_*5 + 255)/256, 256, 0, stream>>>(conv_w5, wp5, 5);
  repack_kernel<<<(CK_*D_*7 + 255)/256, 256, 0, stream>>>(conv_w7, wp7, 7);
  // 1) conv branches -> cnn_out (M x 768)
  conv_kernel<<<M_/16, 128, 0, stream>>>(x, wp3, conv_b3, cnn, 3, 0);
  conv_kernel<<<M_/16, 128, 0, stream>>>(x, wp5, conv_b5, cnn, 5, 256);
  conv_kernel<<<M_/16, 128, 0, stream>>>(x, wp7, conv_b7, cnn, 7, 512);
  // 2) input projections (async-LDS staged A)
  proj_kernel<<<M_/16, 384, 0, stream>>>(cnn, Wxrz_w, Wxrz_b, arz, 768);
  proj_kernel<<<M_/16, 192, 0, stream>>>(cnn, Wxn_w,  Wxn_b,  an,  384);
  // 3) pre-scan LayerNorms (in place)
  ln_rows<<<M_, 256, 0, stream>>>(arz, arz, lnx1_g, lnx1_b, 768);
  ln_rows<<<M_, 256, 0, stream>>>(an,  an,  lnx2_g, lnx2_b, 384);
  // 4) bidirectional GRU scan -> outc (M x 768, fwd|bwd concatenated)
  scan_kernel<<<dim3(B_/16, 2), 384, 0, stream>>>(arz, an, Whrz_w, Whrz_b,
                                                  Whn_w, Whn_b, lnh1_g, lnh1_b,
                                                  lnh2_g, lnh2_b, outc);
  // 5) final LayerNorm -> d_out
  ln_rows<<<M_, 256, 0, stream>>>(outc, out, out_ln_g, out_ln_b, 768);
}